// Attention_8813272891969
// MI455X (gfx1250) — compile-verified
//
#include <hip/hip_runtime.h>
#include <hip/hip_bf16.h>

typedef __attribute__((ext_vector_type(2))) float v2f;
typedef __attribute__((ext_vector_type(8))) float v8f;

#define BB 32
#define TT 4096
#define DIN 512
#define DK 128
#define DV 128
#define NCHUNK 16
#define WAVES 8
#define ROWS_PER_WAVE (TT / NCHUNK / WAVES) /* 32 */

#define SENTINEL (-3.4e38f)
#define GUARD (-1.0e30f)

__device__ __forceinline__ float wave_reduce_add(float v) {
    for (int off = 16; off > 0; off >>= 1) v += __shfl_xor(v, off, 32);
    return v;
}

__device__ __forceinline__ float4 fma4(float4 a, float f, float w, float4 x) {
    float4 r;
    r.x = a.x * f + w * x.x;
    r.y = a.y * f + w * x.y;
    r.z = a.z * f + w * x.z;
    r.w = a.w * f + w * x.w;
    return r;
}

// gfx1250 async copy: per-lane 16B global -> LDS, tracked by ASYNCcnt.
__device__ __forceinline__ void async_b128(unsigned lds_addr, const float* gsrc) {
    asm volatile("global_load_async_to_lds_b128 %0, %1, off"
                 :: "v"(lds_addr), "v"((unsigned long long)(uintptr_t)gsrc)
                 : "memory");
}

// ---------------------------------------------------------------------------
// Kernel 1: wq[b][d] = sum_k Wk[d,k] * q[b,k]
// GEMM: C[512,32] = Wk[512,128] x q^T[128,32] via V_WMMA_F32_16X16X4_F32
// grid = (32 d-tiles, 2 b-tiles), 1 wave per block
// ---------------------------------------------------------------------------
__global__ void wq_wmma_kernel(const float* __restrict__ Wk,
                               const float* __restrict__ q,
                               float* __restrict__ wq) {
    const int mt = blockIdx.x;            // d-tile 0..31
    const int nt = blockIdx.y;            // b-tile 0..1
    const int lane = threadIdx.x;         // 0..31
    const int r16 = lane & 15;
    const int kh = (lane >> 4) << 1;      // lanes 0-15 -> K {0,1}; lanes 16-31 -> K {2,3}

    v8f c = {};
    const float* ap = Wk + (size_t)(mt * 16 + r16) * DK + kh;  // A row (d index)
    const float* bp = q + (size_t)(nt * 16 + r16) * DK + kh;   // B col (b index)
#pragma unroll
    for (int k0 = 0; k0 < DK; k0 += 4) {
        v2f a, b;
        a.x = ap[k0 + 0];
        a.y = ap[k0 + 1];
        b.x = bp[k0 + 0];
        b.y = bp[k0 + 1];
        c = __builtin_amdgcn_wmma_f32_16x16x4_f32(false, a, false, b,
                                                  (short)0, c, false, false);
    }
    // C layout: VGPR r -> lanes 0-15: M=r, N=lane ; lanes 16-31: M=r+8, N=lane-16
    const int mb = (lane >> 4) ? 8 : 0;
#pragma unroll
    for (int r = 0; r < 8; ++r) {
        const int d = mt * 16 + mb + r;   // M index (d)
        const int b = nt * 16 + r16;      // N index (batch)
        wq[(size_t)b * DIN + d] = c[r];
    }
}

// ---------------------------------------------------------------------------
// Kernel 2: qbk[b] = sum_k q[b,k] * bk[k]
// ---------------------------------------------------------------------------
__global__ void qbk_kernel(const float* __restrict__ q,
                           const float* __restrict__ bk,
                           float* __restrict__ qbk) {
    const int b = threadIdx.x; // 32 threads
    float acc = 0.f;
    for (int k = 0; k < DK; ++k) acc += q[b * DK + k] * bk[k];
    qbk[b] = acc;
}

// ---------------------------------------------------------------------------
// Kernel 3: single streaming pass over enc_output (the 268 MB / HBM-bound
// part). Rows are staged global->LDS with gfx1250 async-tensor copies
// (global_load_async_to_lds_b128 + s_wait_asynccnt), double-buffered per
// wave, then consumed via ds_load_b128 into the online-softmax accumulator.
// Writes per-(b,chunk) partial (m, s, acc[512]) after an in-block LDS merge.
// ---------------------------------------------------------------------------
__global__ void stream_softmax_kernel(const float* __restrict__ enc,
                                      const float* __restrict__ wq,
                                      const float* __restrict__ qbk,
                                      const long long* __restrict__ lens,
                                      float* __restrict__ pm,
                                      float* __restrict__ ps,
                                      float* __restrict__ pacc) {
    const int b = blockIdx.x;
    const int chunk = blockIdx.y;
    const int tid = threadIdx.x;
    const int wave = tid >> 5;
    const int lane = tid & 31;
    const int len = (int)lens[b];
    const float qb = qbk[b];
    const float scale = 0.08838834764831845f; // 1/sqrt(128)

    __shared__ __align__(16) float stage[WAVES][2][DIN]; // 32 KB double buffer
    __shared__ float sm[WAVES];
    __shared__ float ss[WAVES];
    __shared__ __align__(16) float sacc[WAVES][DIN];     // 16 KB merge buffer

    const float4* wqv = (const float4*)(wq + (size_t)b * DIN);
    const float4 w0 = wqv[lane], w1 = wqv[lane + 32];
    const float4 w2 = wqv[lane + 64], w3 = wqv[lane + 96];

    float4 a0 = {0.f, 0.f, 0.f, 0.f}, a1 = a0, a2 = a0, a3 = a0;
    float m = SENTINEL;
    float s = 0.f;

    const int row0 = (chunk * WAVES + wave) * ROWS_PER_WAVE;
    int rend = row0 + ROWS_PER_WAVE;
    if (rend > len) rend = len;
    const int nrows = rend - row0; // may be <= 0 for fully masked waves

    // per-lane LDS byte addresses of this wave's two staging buffers
    const unsigned lbase0 =
        (unsigned)(uintptr_t)(&stage[wave][0][0]) + (unsigned)(lane * 16);
    const unsigned lbase1 =
        (unsigned)(uintptr_t)(&stage[wave][1][0]) + (unsigned)(lane * 16);

    if (nrows > 0) {
        // prologue: async-stage row0 into buffer 0
        {
            const float* rp = enc + ((size_t)b * TT + row0) * DIN + 4 * lane;
            async_b128(lbase0 + 0 * 512, rp + 0);
            async_b128(lbase0 + 1 * 512, rp + 128);
            async_b128(lbase0 + 2 * 512, rp + 256);
            async_b128(lbase0 + 3 * 512, rp + 384);
        }
        int buf = 0;
        for (int r = 0; r < nrows; ++r) {
            const int t = row0 + r;
            const bool have_next = (r + 1 < nrows);
            if (have_next) {
                const unsigned lb = buf ? lbase0 : lbase1; // other buffer
                const float* rp = enc + ((size_t)b * TT + t + 1) * DIN + 4 * lane;
                async_b128(lb + 0 * 512, rp + 0);
                async_b128(lb + 1 * 512, rp + 128);
                async_b128(lb + 2 * 512, rp + 256);
                async_b128(lb + 3 * 512, rp + 384);
                // current row's 4 copies retired; next row's 4 still in flight
                asm volatile("s_wait_asynccnt 0x4" ::: "memory");
            } else {
                asm volatile("s_wait_asynccnt 0x0" ::: "memory");
            }

            const float4* sv = (const float4*)(&stage[wave][buf][0]);
            const float4 x0 = sv[lane], x1 = sv[lane + 32];
            const float4 x2 = sv[lane + 64], x3 = sv[lane + 96];

            float d = x0.x * w0.x + x0.y * w0.y + x0.z * w0.z + x0.w * w0.w;
            d += x1.x * w1.x + x1.y * w1.y + x1.z * w1.z + x1.w * w1.w;
            d += x2.x * w2.x + x2.y * w2.y + x2.z * w2.z + x2.w * w2.w;
            d += x3.x * w3.x + x3.y * w3.y + x3.z * w3.z + x3.w * w3.w;
            d = wave_reduce_add(d);

            const float e = (d + qb) * scale;
            const float mnew = fmaxf(m, e);
            const float f = __expf(m - mnew);   // underflows to 0 on first iter
            const float wgt = __expf(e - mnew);
            s = s * f + wgt;
            a0 = fma4(a0, f, wgt, x0);
            a1 = fma4(a1, f, wgt, x1);
            a2 = fma4(a2, f, wgt, x2);
            a3 = fma4(a3, f, wgt, x3);
            m = mnew;
            buf ^= 1;
        }
    }

    // ---- block-level merge of the 8 wave partials via LDS ----
    float4* saccv = (float4*)(&sacc[wave][0]);
    saccv[lane] = a0;
    saccv[lane + 32] = a1;
    saccv[lane + 64] = a2;
    saccv[lane + 96] = a3;
    if (lane == 0) { sm[wave] = m; ss[wave] = s; }
    __syncthreads();

    float M = SENTINEL;
    for (int i = 0; i < WAVES; ++i) M = fmaxf(M, sm[i]);
    float S = 0.f;
    float wv[WAVES];
    for (int i = 0; i < WAVES; ++i) {
        const float wi = (sm[i] > GUARD) ? __expf(sm[i] - M) : 0.f;
        wv[i] = wi;
        S += wi * ss[i];
    }
    const int p = b * NCHUNK + chunk;
    for (int c = tid; c < DIN; c += WAVES * 32) {
        float acc = 0.f;
        for (int i = 0; i < WAVES; ++i) acc += wv[i] * sacc[i][c];
        pacc[(size_t)p * DIN + c] = acc;
    }
    if (tid == 0) { pm[p] = M; ps[p] = S; }
}

// ---------------------------------------------------------------------------
// Kernel 4: flash-style combine of the 16 chunk partials per batch:
// u[b][d] = (sum_i e^{m_i - M} acc_i[d]) / (sum_i e^{m_i - M} s_i)
// ---------------------------------------------------------------------------
__global__ void combine_kernel(const float* __restrict__ pm,
                               const float* __restrict__ ps,
                               const float* __restrict__ pacc,
                               float* __restrict__ u) {
    const int b = blockIdx.x;
    const int tid = threadIdx.x; // 128
    __shared__ float lm[NCHUNK];
    __shared__ float lsw[NCHUNK];
    if (tid < NCHUNK) {
        lm[tid] = pm[b * NCHUNK + tid];
        lsw[tid] = ps[b * NCHUNK + tid];
    }
    __syncthreads();
    float M = SENTINEL;
    for (int i = 0; i < NCHUNK; ++i) M = fmaxf(M, lm[i]);
    float S = 0.f;
    float w[NCHUNK];
    for (int i = 0; i < NCHUNK; ++i) {
        const float wi = (lm[i] > GUARD) ? __expf(lm[i] - M) : 0.f;
        w[i] = wi;
        S += wi * lsw[i];
    }
    const float invS = 1.0f / S;
    for (int c = tid; c < DIN; c += 128) {
        float acc = 0.f;
        for (int i = 0; i < NCHUNK; ++i)
            acc += w[i] * pacc[(size_t)(b * NCHUNK + i) * DIN + c];
        u[(size_t)b * DIN + c] = acc * invS;
    }
}

// ---------------------------------------------------------------------------
// Kernel 5: context[32,128] = u[32,512] @ Wv[512,128] + bv
// via V_WMMA_F32_16X16X4_F32. grid = (2 m-tiles, 8 n-tiles), 1 wave each.
// ---------------------------------------------------------------------------
__global__ void proj_wmma_kernel(const float* __restrict__ u,
                                 const float* __restrict__ Wv,
                                 const float* __restrict__ bv,
                                 float* __restrict__ out) {
    const int mt = blockIdx.x;          // batch tile 0..1
    const int nt = blockIdx.y;          // v-col tile 0..7
    const int lane = threadIdx.x;
    const int r16 = lane & 15;
    const int kh = (lane >> 4) << 1;

    v8f c = {};
    const float* ap = u + (size_t)(mt * 16 + r16) * DIN + kh;
    const float* bp = Wv + (size_t)kh * DV + (nt * 16 + r16);
#pragma unroll
    for (int k0 = 0; k0 < DIN; k0 += 4) {
        v2f a, b;
        a.x = ap[k0 + 0];
        a.y = ap[k0 + 1];
        b.x = bp[(size_t)(k0 + 0) * DV];
        b.y = bp[(size_t)(k0 + 1) * DV];
        c = __builtin_amdgcn_wmma_f32_16x16x4_f32(false, a, false, b,
                                                  (short)0, c, false, false);
    }
    const float bias = bv[nt * 16 + r16];
    const int mb = (lane >> 4) ? 8 : 0;
#pragma unroll
    for (int r = 0; r < 8; ++r) {
        const int m = mt * 16 + mb + r;
        const int n = nt * 16 + r16;
        out[(size_t)m * DV + n] = c[r] + bias;
    }
}

// ---------------------------------------------------------------------------
extern "C" void kernel_launch(void* const* d_in, const int* in_sizes, int n_in,
                              void* d_out, int out_size, void* d_ws, size_t ws_size,
                              hipStream_t stream) {
    const float* enc = (const float*)d_in[0];
    const float* q = (const float*)d_in[1];
    const long long* lens = (const long long*)d_in[2];
    const float* Wk = (const float*)d_in[3];
    const float* bk = (const float*)d_in[4];
    const float* Wv = (const float*)d_in[5];
    const float* bv = (const float*)d_in[6];
    float* out = (float*)d_out;

    float* ws = (float*)d_ws;
    float* wq = ws;                              // 32*512
    float* qbk = wq + BB * DIN;                  // 32
    float* pm = qbk + BB;                        // 32*16
    float* ps = pm + BB * NCHUNK;                // 32*16
    float* pacc = ps + BB * NCHUNK;              // 32*16*512
    float* u = pacc + (size_t)BB * NCHUNK * DIN; // 32*512
    // total ~1.2 MB of f32 workspace

    wq_wmma_kernel<<<dim3(DIN / 16, BB / 16), 32, 0, stream>>>(Wk, q, wq);
    qbk_kernel<<<1, BB, 0, stream>>>(q, bk, qbk);
    stream_softmax_kernel<<<dim3(BB, NCHUNK), WAVES * 32, 0, stream>>>(
        enc, wq, qbk, lens, pm, ps, pacc);
    combine_kernel<<<BB, 128, 0, stream>>>(pm, ps, pacc, u);
    proj_wmma_kernel<<<dim3(BB / 16, DV / 16), 32, 0, stream>>>(u, Wv, bv, out);
}